// VolumeRenderer_1649267442059
// MI455X (gfx1250) — compile-verified
//
#include <hip/hip_runtime.h>
#include <math.h>

#define NG   2048
#define H_   128
#define W_   128
#define D_   32
#define F_   2
#define LOG2E 1.4426950408889634f

typedef float v2f __attribute__((ext_vector_type(2)));
typedef float v8f __attribute__((ext_vector_type(8)));

// ---------------------------------------------------------------------------
// Kernel 1: per-Gaussian preprocessing.
// Packs 16 floats per Gaussian:
//   P[4n+0] = {mu_x, mu_y, mu_z, cov02}
//   P[4n+1] = {cov12, zinv, 3*zstd, opacity}
//   P[4n+2] = {-0.5*log2e*pa, -log2e*pb, -0.5*log2e*pc, feat0}
//   P[4n+3] = {feat1, 0, 0, 0}
// ---------------------------------------------------------------------------
__global__ __launch_bounds__(256) void prep_gaussians(
    const float* __restrict__ xyz, const float* __restrict__ scale,
    const float* __restrict__ rot, const float* __restrict__ opac,
    const float* __restrict__ feat, float4* __restrict__ P)
{
    int n = blockIdx.x * 256 + threadIdx.x;
    if (n >= NG) return;

    float qw = rot[n*4+0], qx = rot[n*4+1], qy = rot[n*4+2], qz = rot[n*4+3];
    float inv = 1.0f / (sqrtf(qw*qw + qx*qx + qy*qy + qz*qz) + 1e-8f);
    qw *= inv; qx *= inv; qy *= inv; qz *= inv;

    float r00 = 1.f - 2.f*(qy*qy + qz*qz);
    float r01 = 2.f*(qx*qy - qw*qz);
    float r02 = 2.f*(qx*qz + qw*qy);
    float r10 = 2.f*(qx*qy + qw*qz);
    float r11 = 1.f - 2.f*(qx*qx + qz*qz);
    float r12 = 2.f*(qy*qz - qw*qx);
    float r20 = 2.f*(qx*qz - qw*qy);
    float r21 = 2.f*(qy*qz + qw*qx);
    float r22 = 1.f - 2.f*(qx*qx + qy*qy);

    float sx = scale[n*3+0], sy = scale[n*3+1], sz = scale[n*3+2];
    float m00 = r00*sx, m01 = r01*sy, m02 = r02*sz;
    float m10 = r10*sx, m11 = r11*sy, m12 = r12*sz;
    float m20 = r20*sx, m21 = r21*sy, m22 = r22*sz;

    float cov00 = m00*m00 + m01*m01 + m02*m02;
    float cov01 = m00*m10 + m01*m11 + m02*m12;
    float cov02 = m00*m20 + m01*m21 + m02*m22;
    float cov11 = m10*m10 + m11*m11 + m12*m12;
    float cov12 = m10*m20 + m11*m21 + m12*m22;
    float cov22 = m20*m20 + m21*m21 + m22*m22;

    float zv   = cov22 + 1e-8f;
    float zinv = 1.0f / zv;
    float thr3 = 3.0f * sqrtf(zv);

    float a  = cov00 - zinv*cov02*cov02 + 1e-6f;
    float b  = cov01 - zinv*cov02*cov12;
    float cc = cov11 - zinv*cov12*cov12 + 1e-6f;
    float idet = 1.0f / (a*cc - b*b);
    float pa = cc*idet, pb = -b*idet, pc = a*idet;

    P[n*4+0] = make_float4(xyz[n*3+0], xyz[n*3+1], xyz[n*3+2], cov02);
    P[n*4+1] = make_float4(cov12, zinv, thr3, opac[n]);
    P[n*4+2] = make_float4(-0.5f*LOG2E*pa, -LOG2E*pb, -0.5f*LOG2E*pc, feat[n*2+0]);
    P[n*4+3] = make_float4(feat[n*2+1], 0.f, 0.f, 0.f);
}

// ---------------------------------------------------------------------------
// Kernel 2: render one (slice d, image row h) per block; 8 waves x 16 pixels.
// Each wave accumulates a 16(pixels) x 16(feature cols; 2 used) f32 tile via
// V_WMMA_F32_16X16X4_F32, folding 4 Gaussians per WMMA.
// A layout (16x4 f32): lanes 0-15 -> K0,K1 ; lanes 16-31 -> K2,K3.
// ---------------------------------------------------------------------------
__global__ __launch_bounds__(256) void render_slices(
    const float4* __restrict__ P, float* __restrict__ out)
{
    __shared__ float4 s0[256];   // {mux, muy, op*mask, feat0}
    __shared__ float4 s1[256];   // {ea, eb, ec, feat1}

    const int d      = blockIdx.x;
    const int h      = blockIdx.y;
    const int tid    = threadIdx.x;
    const int lane   = tid & 31;
    const int wave   = tid >> 5;
    const int laneN  = lane & 15;
    const int laneHi = lane >> 4;           // 0: K0/K1, 1: K2/K3
    const int wbase  = wave * 16;

    const float z  = (float)d * (1.0f / (D_ - 1));
    const float yh = (float)h * (1.0f / (H_ - 1));
    const float xw = (float)(wbase + laneN) * (1.0f / (W_ - 1));

    v8f acc = {0.f, 0.f, 0.f, 0.f, 0.f, 0.f, 0.f, 0.f};

    for (int c0 = 0; c0 < NG; c0 += 256) {
        // Prefetch next chunk's params into cache (global_prefetch_b8).
        int np = c0 + 256;
        if (np < NG)
            __builtin_prefetch((const void*)&P[(size_t)(np + tid) * 4], 0, 3);

        // Slice-specialize 256 Gaussians into LDS (one per thread).
        {
            int g = c0 + tid;
            float4 q0 = P[g*4+0];
            float4 q1 = P[g*4+1];
            float4 q2 = P[g*4+2];
            float4 q3 = P[g*4+3];
            float tt  = q1.y * (z - q0.z);            // zinv*(z - mu_z)
            float mux = q0.x + q0.w * tt;
            float muy = q0.y + q1.x * tt;
            float msk = (fabsf(q0.z - z) < q1.z) ? 1.0f : 0.0f;
            s0[tid] = make_float4(mux, muy, q1.w * msk, q2.w);
            s1[tid] = make_float4(q2.x, q2.y, q2.z, q3.x);
        }
        __syncthreads();

        // 64 WMMA steps of 4 Gaussians each.
        for (int s = 0; s < 64; ++s) {
            int gA = (s << 2) + (laneHi << 1);        // this lane's first Gaussian
            float4 a0 = s0[gA];
            float4 b0 = s0[gA + 1];
            bool live = (a0.z != 0.f) || (b0.z != 0.f);
            if (__any(live)) {                        // wave-uniform skip; EXEC stays full
                float4 a1 = s1[gA];
                float4 b1 = s1[gA + 1];

                float dxA = xw - a0.x, dyA = yh - a0.y;
                float mA  = dxA * (a1.x * dxA + a1.y * dyA) + a1.z * dyA * dyA;
                // Raw hardware exp2 (v_exp_f32): skips the subnormal-range fixup
                // sequence; alphas below 2^-126 are numerically irrelevant here.
                float alA = a0.z * __builtin_amdgcn_exp2f(mA);

                float dxB = xw - b0.x, dyB = yh - b0.y;
                float mB  = dxB * (b1.x * dxB + b1.y * dyB) + b1.z * dyB * dyB;
                float alB = b0.z * __builtin_amdgcn_exp2f(mB);

                v2f Am = {alA, alB};
                // B (4x16): column 0 = feat0, column 1 = feat1, rest 0.
                float e0 = (laneN == 0) ? a0.w : (laneN == 1) ? a1.w : 0.f;
                float e1 = (laneN == 0) ? b0.w : (laneN == 1) ? b1.w : 0.f;
                v2f Bm = {e0, e1};

                acc = __builtin_amdgcn_wmma_f32_16x16x4_f32(
                          false, Am, false, Bm, (short)0, acc, false, false);
            }
        }
        __syncthreads();
    }

    // D layout: VGPR r, lanes 0-15 -> (M=r, N=lane), lanes 16-31 -> (M=r+8, N=lane-16).
    // Output index (H,W,D,F): ((h*W + w)*D + d)*F + f, with w = wbase+M, f = laneN.
    if (laneN < F_) {
        #pragma unroll
        for (int r = 0; r < 8; ++r) {
            int M = r + (laneHi << 3);
            int w = wbase + M;
            out[(((size_t)h * W_ + w) * D_ + d) * F_ + laneN] = acc[r];
        }
    }
}

// ---------------------------------------------------------------------------
extern "C" void kernel_launch(void* const* d_in, const int* in_sizes, int n_in,
                              void* d_out, int out_size, void* d_ws, size_t ws_size,
                              hipStream_t stream)
{
    const float* xyz   = (const float*)d_in[0];
    const float* scale = (const float*)d_in[1];
    const float* rot   = (const float*)d_in[2];
    const float* opac  = (const float*)d_in[3];
    const float* feat  = (const float*)d_in[4];
    float4* P = (float4*)d_ws;               // 2048 * 64 B = 128 KB scratch

    prep_gaussians<<<NG / 256, 256, 0, stream>>>(xyz, scale, rot, opac, feat, P);
    render_slices<<<dim3(D_, H_), 256, 0, stream>>>(P, (float*)d_out);
}